// SAGERegression_3504693313561
// MI455X (gfx1250) — compile-verified
//
#include <hip/hip_runtime.h>
#include <hip/hip_bf16.h>
#include <stddef.h>

// ---------------------------------------------------------------------------
// SAGE regression for MI455X (gfx1250, wave32).
// Edge phase: 1 thread/edge, float4 gather, f32 L2 atomics (features fit L2).
// Node phase: V_WMMA_F32_16X16X4_F32, one wave per 16-node tile, fused
//             mean@Wl + x@Wr + bias -> relu -> @Wlin + bias.
// ---------------------------------------------------------------------------

typedef float v2f __attribute__((ext_vector_type(2)));
typedef float v8f __attribute__((ext_vector_type(8)));

#define HID 16
#define LDS_PITCH 18   // 16 + 2 pad: keeps float2 LDS loads 8B aligned, breaks bank patterns

__device__ __forceinline__ v8f wmma_f32_k4(v2f a, v2f b, v8f c) {
    // D(16x16,f32) = A(16x4,f32) x B(4x16,f32) + C
    return __builtin_amdgcn_wmma_f32_16x16x4_f32(
        /*neg_a=*/false, a, /*neg_b=*/false, b,
        /*c_mod=*/(short)0, c, /*reuse_a=*/false, /*reuse_b=*/false);
}

// ---------------------------------------------------------------------------
// zero fill (grid-stride)
// ---------------------------------------------------------------------------
__global__ void sage_fill0(float* __restrict__ p, size_t n) {
    size_t i = (size_t)blockIdx.x * blockDim.x + threadIdx.x;
    size_t stride = (size_t)gridDim.x * blockDim.x;
    for (; i < n; i += stride) p[i] = 0.0f;
}

// ---------------------------------------------------------------------------
// layer 0 edge pass: deg[dst] += 1 (reused by all layers), sum0[dst] += x0[src]
// ---------------------------------------------------------------------------
__global__ void sage_edge_pass0(const float* __restrict__ x0,
                                const int* __restrict__ src,
                                const int* __restrict__ dst,
                                float* __restrict__ deg,
                                float* __restrict__ sum0, int E) {
    int e = blockIdx.x * blockDim.x + threadIdx.x;
    if (e >= E) return;
    int s = src[e], d = dst[e];
    atomicAdd(&deg[d], 1.0f);
    atomicAdd(&sum0[d], x0[s]);
}

// ---------------------------------------------------------------------------
// layers 1/2 edge pass: summed[dst][0:16] += x[src][0:16]
// ---------------------------------------------------------------------------
__global__ void sage_scatter16(const float* __restrict__ x,
                               const int* __restrict__ src,
                               const int* __restrict__ dst,
                               float* __restrict__ summed, int E) {
    int e = blockIdx.x * blockDim.x + threadIdx.x;
    if (e >= E) return;
    int s = src[e], d = dst[e];
    const float4* xr = (const float4*)(x + (size_t)s * HID);
    float4 v0 = xr[0], v1 = xr[1], v2 = xr[2], v3 = xr[3];
    float* o = summed + (size_t)d * HID;
    atomicAdd(o + 0,  v0.x); atomicAdd(o + 1,  v0.y);
    atomicAdd(o + 2,  v0.z); atomicAdd(o + 3,  v0.w);
    atomicAdd(o + 4,  v1.x); atomicAdd(o + 5,  v1.y);
    atomicAdd(o + 6,  v1.z); atomicAdd(o + 7,  v1.w);
    atomicAdd(o + 8,  v2.x); atomicAdd(o + 9,  v2.y);
    atomicAdd(o + 10, v2.z); atomicAdd(o + 11, v2.w);
    atomicAdd(o + 12, v3.x); atomicAdd(o + 13, v3.y);
    atomicAdd(o + 14, v3.z); atomicAdd(o + 15, v3.w);
}

// ---------------------------------------------------------------------------
// layer 0 node update (din=1): h = relu(mean*Wl0 + x*Wr0 + bl0); out = h@Wlin0 + blin0
// 4 waves/block, 1 tile (16 nodes) per wave. Requires N % 64 == 0.
// ---------------------------------------------------------------------------
__global__ void sage_update0(const float* __restrict__ x0,
                             const float* __restrict__ sum0,
                             const float* __restrict__ deg,
                             const float* __restrict__ Wl0,  // [16]
                             const float* __restrict__ bl0,  // [16]
                             const float* __restrict__ Wr0,  // [16]
                             const float* __restrict__ Wlin, // [16,16]
                             const float* __restrict__ blin, // [16]
                             float* __restrict__ xout) {
    __shared__ float lds[4][16 * LDS_PITCH];
    const int wave = threadIdx.x >> 5;
    const int lane = threadIdx.x & 31;
    const int nb   = (blockIdx.x * 4 + wave) * 16;   // tile base node
    const int n    = lane & 15;                       // column
    const int kh   = lane >> 4;                       // K/M half selector
    float* L = lds[wave];

    // build H tile (C-layout coverage: rows r+8*kh, col n)
    const float wl = Wl0[n], wr = Wr0[n], bb = bl0[n];
    #pragma unroll
    for (int r = 0; r < 8; ++r) {
        int m = r + 8 * kh;
        int node = nb + m;
        float mean = sum0[node] / fmaxf(deg[node], 1.0f);
        L[m * LDS_PITCH + n] = fmaxf(mean * wl + x0[node] * wr + bb, 0.0f);
    }
    __syncthreads();

    // out = H @ Wlin + blin via 4x WMMA f32 K=4
    v8f c = {0.f,0.f,0.f,0.f,0.f,0.f,0.f,0.f};
    #pragma unroll
    for (int kc = 0; kc < 4; ++kc) {
        int k0 = 4 * kc + 2 * kh;
        v2f a; a.x = L[n * LDS_PITCH + k0]; a.y = L[n * LDS_PITCH + k0 + 1];
        v2f b; b.x = Wlin[k0 * 16 + n];     b.y = Wlin[(k0 + 1) * 16 + n];
        c = wmma_f32_k4(a, b, c);
    }
    const float bn = blin[n];
    #pragma unroll
    for (int r = 0; r < 8; ++r)
        xout[(size_t)(nb + r + 8 * kh) * HID + n] = c[r] + bn;
}

// ---------------------------------------------------------------------------
// layers 1/2 node update (din=16):
//   out = (relu( (summed/deg)@Wl + x@Wr + bl )) @ Wlin + blin
// ---------------------------------------------------------------------------
__global__ void sage_update16(const float* __restrict__ xin,
                              const float* __restrict__ summed,
                              const float* __restrict__ deg,
                              const float* __restrict__ Wl,
                              const float* __restrict__ bl,
                              const float* __restrict__ Wr,
                              const float* __restrict__ Wlin,
                              const float* __restrict__ blin,
                              float* __restrict__ xout) {
    __shared__ float lds[4][16 * LDS_PITCH];
    const int wave = threadIdx.x >> 5;
    const int lane = threadIdx.x & 31;
    const int nb   = (blockIdx.x * 4 + wave) * 16;
    const int n    = lane & 15;
    const int kh   = lane >> 4;
    float* L = lds[wave];

    const float* xrow = xin    + (size_t)(nb + n) * HID;   // A-row for this lane
    const float* srow = summed + (size_t)(nb + n) * HID;
    const float  invd = 1.0f / fmaxf(deg[nb + n], 1.0f);

    v8f c = {0.f,0.f,0.f,0.f,0.f,0.f,0.f,0.f};
    #pragma unroll
    for (int kc = 0; kc < 4; ++kc) {                 // mean @ Wl
        int k0 = 4 * kc + 2 * kh;
        v2f a; a.x = srow[k0] * invd; a.y = srow[k0 + 1] * invd;
        v2f b; b.x = Wl[k0 * 16 + n]; b.y = Wl[(k0 + 1) * 16 + n];
        c = wmma_f32_k4(a, b, c);
    }
    #pragma unroll
    for (int kc = 0; kc < 4; ++kc) {                 // + x @ Wr
        int k0 = 4 * kc + 2 * kh;
        v2f a; a.x = xrow[k0];        a.y = xrow[k0 + 1];
        v2f b; b.x = Wr[k0 * 16 + n]; b.y = Wr[(k0 + 1) * 16 + n];
        c = wmma_f32_k4(a, b, c);
    }
    const float bln = bl[n];
    #pragma unroll
    for (int r = 0; r < 8; ++r)                      // bias + relu -> LDS (C-layout)
        L[(r + 8 * kh) * LDS_PITCH + n] = fmaxf(c[r] + bln, 0.0f);
    __syncthreads();

    v8f c2 = {0.f,0.f,0.f,0.f,0.f,0.f,0.f,0.f};
    #pragma unroll
    for (int kc = 0; kc < 4; ++kc) {                 // @ Wlin
        int k0 = 4 * kc + 2 * kh;
        v2f a; a.x = L[n * LDS_PITCH + k0]; a.y = L[n * LDS_PITCH + k0 + 1];
        v2f b; b.x = Wlin[k0 * 16 + n];     b.y = Wlin[(k0 + 1) * 16 + n];
        c2 = wmma_f32_k4(a, b, c2);
    }
    const float bn = blin[n];
    #pragma unroll
    for (int r = 0; r < 8; ++r)
        xout[(size_t)(nb + r + 8 * kh) * HID + n] = c2[r] + bn;
}

// ---------------------------------------------------------------------------
// attention pool, pass 1: denom[batch[n]][j] += exp(alpha*x[n][j])
// ---------------------------------------------------------------------------
__global__ void sage_pool1(const float* __restrict__ x,
                           const int* __restrict__ batch,
                           float* __restrict__ denom, int N) {
    int i = blockIdx.x * blockDim.x + threadIdx.x;
    if (i >= N) return;
    int b = batch[i];
    const float* xr = x + (size_t)i * HID;
    float* dr = denom + (size_t)b * HID;
    #pragma unroll
    for (int j = 0; j < HID; ++j)
        atomicAdd(dr + j, expf(0.2f * xr[j]));
}

// pass 2: pooled[b][j] += x*score/denom
__global__ void sage_pool2(const float* __restrict__ x,
                           const int* __restrict__ batch,
                           const float* __restrict__ denom,
                           float* __restrict__ pooled, int N) {
    int i = blockIdx.x * blockDim.x + threadIdx.x;
    if (i >= N) return;
    int b = batch[i];
    const float* xr = x + (size_t)i * HID;
    const float* dr = denom + (size_t)b * HID;
    float* pr = pooled + (size_t)b * HID;
    #pragma unroll
    for (int j = 0; j < HID; ++j) {
        float s = expf(0.2f * xr[j]);
        atomicAdd(pr + j, xr[j] * s / dr[j]);
    }
}

// ---------------------------------------------------------------------------
// MLP head: out = relu(relu(pooled@Wo0+bo0)@Wo1+bo1)@Wo2 + bo2
// one wave (32 threads) per 16 graphs; 64 blocks total.
// ---------------------------------------------------------------------------
__global__ void sage_head(const float* __restrict__ pooled,
                          const float* __restrict__ Wo0, const float* __restrict__ bo0,
                          const float* __restrict__ Wo1, const float* __restrict__ bo1,
                          const float* __restrict__ Wo2, const float* __restrict__ bo2,
                          float* __restrict__ out) {
    __shared__ float h1[16 * LDS_PITCH];
    __shared__ float h2[16 * LDS_PITCH];
    const int lane = threadIdx.x;
    const int gb   = blockIdx.x * 16;
    const int n    = lane & 15;
    const int kh   = lane >> 4;

    const float* prow = pooled + (size_t)(gb + n) * HID;
    v8f c = {0.f,0.f,0.f,0.f,0.f,0.f,0.f,0.f};
    #pragma unroll
    for (int kc = 0; kc < 4; ++kc) {
        int k0 = 4 * kc + 2 * kh;
        v2f a; a.x = prow[k0];         a.y = prow[k0 + 1];
        v2f b; b.x = Wo0[k0 * 16 + n]; b.y = Wo0[(k0 + 1) * 16 + n];
        c = wmma_f32_k4(a, b, c);
    }
    float b0 = bo0[n];
    #pragma unroll
    for (int r = 0; r < 8; ++r)
        h1[(r + 8 * kh) * LDS_PITCH + n] = fmaxf(c[r] + b0, 0.0f);
    __syncthreads();

    v8f c2 = {0.f,0.f,0.f,0.f,0.f,0.f,0.f,0.f};
    #pragma unroll
    for (int kc = 0; kc < 4; ++kc) {
        int k0 = 4 * kc + 2 * kh;
        v2f a; a.x = h1[n * LDS_PITCH + k0]; a.y = h1[n * LDS_PITCH + k0 + 1];
        v2f b; b.x = Wo1[k0 * 16 + n];       b.y = Wo1[(k0 + 1) * 16 + n];
        c2 = wmma_f32_k4(a, b, c2);
    }
    float b1 = bo1[n];
    #pragma unroll
    for (int r = 0; r < 8; ++r)
        h2[(r + 8 * kh) * LDS_PITCH + n] = fmaxf(c2[r] + b1, 0.0f);
    __syncthreads();

    if (lane < 16) {
        float acc = bo2[0];
        #pragma unroll
        for (int j = 0; j < HID; ++j)
            acc += h2[lane * LDS_PITCH + j] * Wo2[j];
        out[gb + lane] = acc;
    }
}

// ---------------------------------------------------------------------------
// launcher
// ---------------------------------------------------------------------------
extern "C" void kernel_launch(void* const* d_in, const int* in_sizes, int n_in,
                              void* d_out, int out_size, void* d_ws, size_t ws_size,
                              hipStream_t stream) {
    const float* x0    = (const float*)d_in[0];
    const int*   ei    = (const int*)d_in[1];
    const int*   batch = (const int*)d_in[2];
    const float* Wl0 = (const float*)d_in[3],  *bl0 = (const float*)d_in[4];
    const float* Wr0 = (const float*)d_in[5];
    const float* Wlin0 = (const float*)d_in[6], *blin0 = (const float*)d_in[7];
    const float* Wl1 = (const float*)d_in[8],  *bl1 = (const float*)d_in[9];
    const float* Wr1 = (const float*)d_in[10];
    const float* Wlin1 = (const float*)d_in[11], *blin1 = (const float*)d_in[12];
    const float* Wl2 = (const float*)d_in[13], *bl2 = (const float*)d_in[14];
    const float* Wr2 = (const float*)d_in[15];
    const float* Wlin2 = (const float*)d_in[16], *blin2 = (const float*)d_in[17];
    const float* Wo0 = (const float*)d_in[18], *bo0 = (const float*)d_in[19];
    const float* Wo1 = (const float*)d_in[20], *bo1 = (const float*)d_in[21];
    const float* Wo2 = (const float*)d_in[22], *bo2 = (const float*)d_in[23];
    float* out = (float*)d_out;

    const int N = in_sizes[0];          // 200000 (multiple of 64)
    const int E = in_sizes[1] / 2;      // 5,000,000
    const int B = out_size;             // 1024

    // workspace layout (floats): xA | xB | summed | deg | denom | pooled
    float* xA     = (float*)d_ws;
    float* xB     = xA + (size_t)N * HID;
    float* summed = xB + (size_t)N * HID;   // first N floats double as sum0
    float* deg    = summed + (size_t)N * HID;
    float* denom  = deg + N;
    float* pooled = denom + (size_t)B * HID;

    const int eb = (E + 255) / 256;     // edge blocks
    const int tb = (N / 16) / 4;        // tile blocks: 4 waves of 1 tile each

    // zero summed (incl. sum0), deg, denom, pooled in one sweep
    size_t zc = (size_t)N * HID + (size_t)N + 2 * (size_t)B * HID;
    sage_fill0<<<2048, 256, 0, stream>>>(summed, zc);

    // ---- layer 0 ----
    sage_edge_pass0<<<eb, 256, 0, stream>>>(x0, ei, ei + E, deg, summed, E);
    sage_update0<<<tb, 128, 0, stream>>>(x0, summed, deg, Wl0, bl0, Wr0,
                                         Wlin0, blin0, xA);
    // ---- layer 1 ----
    sage_fill0<<<2048, 256, 0, stream>>>(summed, (size_t)N * HID);
    sage_scatter16<<<eb, 256, 0, stream>>>(xA, ei, ei + E, summed, E);
    sage_update16<<<tb, 128, 0, stream>>>(xA, summed, deg, Wl1, bl1, Wr1,
                                          Wlin1, blin1, xB);
    // ---- layer 2 ----
    sage_fill0<<<2048, 256, 0, stream>>>(summed, (size_t)N * HID);
    sage_scatter16<<<eb, 256, 0, stream>>>(xB, ei, ei + E, summed, E);
    sage_update16<<<tb, 128, 0, stream>>>(xB, summed, deg, Wl2, bl2, Wr2,
                                          Wlin2, blin2, xA);
    // ---- attention pool ----
    const int nb2 = (N + 255) / 256;
    sage_pool1<<<nb2, 256, 0, stream>>>(xA, batch, denom, N);
    sage_pool2<<<nb2, 256, 0, stream>>>(xA, batch, denom, pooled, N);
    // ---- MLP head ----
    sage_head<<<B / 16, 32, 0, stream>>>(pooled, Wo0, bo0, Wo1, bo1, Wo2, bo2, out);
}